// PerLayerKVCache_54288386621472
// MI455X (gfx1250) — compile-verified
//
#include <hip/hip_runtime.h>
#include <stdint.h>

// KV-cache scatter update for MI455X (gfx1250).
// Zero-FLOP, pure-bandwidth op -> the CDNA5 feature that applies is the async
// Global<->LDS DMA path (GLOBAL_LOAD_ASYNC_TO_LDS_B128 /
// GLOBAL_STORE_ASYNC_FROM_LDS_B128, ASYNCcnt). One pass over the 512MB output,
// sourcing each (b,s) row from either the val tensors (if scattered there this
// step) or the cache tensors, staged through per-lane LDS slots. NT temporal
// hints on the global side: every byte is single-touch, working set >> 192MB L2.

constexpr int BS     = 32;
constexpr int SL     = 1024;
constexpr int MAX_SL = 2048;
constexpr int NH     = 8;
constexpr int HD     = 128;
constexpr int ROW    = NH * HD;        // 1024 floats per (b, s) row
constexpr size_t CACHE_ELEMS = (size_t)BS * MAX_SL * ROW;  // 67,108,864 f32

// ---- Kernel 1: rev[b*MAX_SL + s] = -1 for all rows (256 KB in d_ws) ----
__global__ __launch_bounds__(256) void kv_init_rev(int* __restrict__ rev) {
    int i = blockIdx.x * 256 + threadIdx.x;
    if (i < BS * MAX_SL) rev[i] = -1;
}

// ---- Kernel 2: rev[b*MAX_SL + input_pos[b][j]] = j ----
// input_pos has distinct indices per row (per reference), so last-writer-wins
// never triggers and the result is deterministic.
__global__ __launch_bounds__(256) void kv_build_rev(const int* __restrict__ pos,
                                                    int* __restrict__ rev) {
    int i = blockIdx.x * 256 + threadIdx.x;     // i in [0, BS*SL)
    if (i >= BS * SL) return;
    int b = i / SL;
    int j = i - b * SL;
    int p = pos[i];                             // 0 <= p < MAX_SL
    rev[b * MAX_SL + p] = j;
}

// ---- Kernel 3: one block per (b, s) output row ----
// A row is 8*128 = 1024 floats = 4KB; 256 lanes x 16B covers it exactly.
// Each lane owns a private 16B LDS slot for K and one for V:
//   async load  global(src) -> LDS   (ASYNCcnt++)
//   s_wait_asynccnt 0                (LDS slots now valid)
//   async store LDS -> global(out)   (ASYNCcnt++)
//   s_wait_asynccnt 0                (stores issued to memory hierarchy)
// No DS barrier needed: lanes never touch each other's slots.
__global__ __launch_bounds__(256) void kv_update(const float* __restrict__ k_cache,
                                                 const float* __restrict__ v_cache,
                                                 const float* __restrict__ k_val,
                                                 const float* __restrict__ v_val,
                                                 const int*   __restrict__ rev,
                                                 float*       __restrict__ out) {
    __shared__ float lds_k[ROW];   // 4 KB
    __shared__ float lds_v[ROW];   // 4 KB

    const int r = blockIdx.x;                   // row index in [0, BS*MAX_SL)
    const int b = r / MAX_SL;
    const int t = threadIdx.x;                  // [0, 256)
    const int j = rev[r];                       // block-uniform scalar load

    const size_t dst = (size_t)r * ROW + (size_t)t * 4;

    const float* ksrc;
    const float* vsrc;
    if (j >= 0) {
        // Row scattered this step: pull from the val tensors; the cache bytes
        // for this row are dead and never read (saves 256MB vs copy+scatter).
        const size_t src = ((size_t)b * SL + (size_t)j) * ROW + (size_t)t * 4;
        ksrc = k_val + src;
        vsrc = v_val + src;
    } else {
        // Untouched row: stream it through from the cache.
        ksrc = k_cache + dst;
        vsrc = v_cache + dst;
    }

    // addrspace(3) pointers are byte offsets from LDS base -> direct cast.
    uint32_t lk = (uint32_t)(uintptr_t)(&lds_k[t * 4]);
    uint32_t lv = (uint32_t)(uintptr_t)(&lds_v[t * 4]);

    asm volatile("global_load_async_to_lds_b128 %0, %1, off th:TH_LOAD_NT"
                 :: "v"(lk), "v"(ksrc) : "memory");
    asm volatile("global_load_async_to_lds_b128 %0, %1, off th:TH_LOAD_NT"
                 :: "v"(lv), "v"(vsrc) : "memory");
    asm volatile("s_wait_asynccnt 0x0" ::: "memory");
    asm volatile("global_store_async_from_lds_b128 %0, %1, off th:TH_STORE_NT"
                 :: "v"(out + dst), "v"(lk) : "memory");
    asm volatile("global_store_async_from_lds_b128 %0, %1, off th:TH_STORE_NT"
                 :: "v"(out + CACHE_ELEMS + dst), "v"(lv) : "memory");
    asm volatile("s_wait_asynccnt 0x0" ::: "memory");
}

extern "C" void kernel_launch(void* const* d_in, const int* in_sizes, int n_in,
                              void* d_out, int out_size, void* d_ws, size_t ws_size,
                              hipStream_t stream) {
    const float* k_cache = (const float*)d_in[0];
    const float* v_cache = (const float*)d_in[1];
    const float* k_val   = (const float*)d_in[2];
    const float* v_val   = (const float*)d_in[3];
    const int*   pos     = (const int*)d_in[4];   // (BS, SL) positions

    int*   rev = (int*)d_ws;                      // BS*MAX_SL ints = 256 KB
    float* out = (float*)d_out;                   // k_out || v_out

    (void)in_sizes; (void)n_in; (void)out_size; (void)ws_size;

    kv_init_rev <<<(BS * MAX_SL + 255) / 256, 256, 0, stream>>>(rev);
    kv_build_rev<<<(BS * SL + 255) / 256,     256, 0, stream>>>(pos, rev);
    kv_update   <<<BS * MAX_SL,               256, 0, stream>>>(
        k_cache, v_cache, k_val, v_val, rev, out);
}